// LaplacianLoss_48808008352295
// MI455X (gfx1250) — compile-verified
//
#include <hip/hip_runtime.h>

// Per-lane 2×f32 A/B operands and 8×f32 C/D accumulator for
// V_WMMA_F32_16X16X4_F32 (16x4 A, 4x16 B, 16x16 C/D over a wave32).
typedef __attribute__((ext_vector_type(2))) float v2f;
typedef __attribute__((ext_vector_type(8))) float v8f;

#define EDGE_LENGTH_SCALE 1.0f

// ---------------------------------------------------------------------------
// Kernel 1: face-parallel gather + scatter.
// acc layout (SoA, 5*N floats): [0,N) tot_x | [N,2N) tot_y | [2N,3N) tot_z |
//                               [3N,4N) edge_sum | [4N,5N) deg (==counts==el_cnt)
// Per face f with vertices (i0,i1,i2):
//   tot[i0] += v1+v2, tot[i1] += v0+v2, tot[i2] += v0+v1
//   edge[i0] += e1+e2, edge[i1] += e1+e3, edge[i2] += e2+e3
//   deg[i*]  += 2
// ---------------------------------------------------------------------------
__global__ void lap_face_scatter(const float* __restrict__ verts,
                                 const long long* __restrict__ idx,
                                 float* __restrict__ acc,
                                 int N, int F) {
    int f = blockIdx.x * blockDim.x + threadIdx.x;
    if (f >= F) return;

    int i0 = (int)idx[3 * (long long)f + 0];
    int i1 = (int)idx[3 * (long long)f + 1];
    int i2 = (int)idx[3 * (long long)f + 2];

    const float* p0 = verts + 3 * (long long)i0;
    const float* p1 = verts + 3 * (long long)i1;
    const float* p2 = verts + 3 * (long long)i2;
    float x0 = p0[0], y0 = p0[1], z0 = p0[2];
    float x1 = p1[0], y1 = p1[1], z1 = p1[2];
    float x2 = p2[0], y2 = p2[1], z2 = p2[2];

    float d10x = x1 - x0, d10y = y1 - y0, d10z = z1 - z0;
    float d20x = x2 - x0, d20y = y2 - y0, d20z = z2 - z0;
    float d21x = x2 - x1, d21y = y2 - y1, d21z = z2 - z1;
    float e1 = sqrtf(d10x * d10x + d10y * d10y + d10z * d10z);
    float e2 = sqrtf(d20x * d20x + d20y * d20y + d20z * d20z);
    float e3 = sqrtf(d21x * d21x + d21y * d21y + d21z * d21z);

    float* tx = acc;
    float* ty = acc + (size_t)N;
    float* tz = acc + (size_t)2 * N;
    float* el = acc + (size_t)3 * N;
    float* dg = acc + (size_t)4 * N;

    // vertex i0: neighbors v1, v2; edges e1, e2
    atomicAdd(&tx[i0], x1 + x2);
    atomicAdd(&ty[i0], y1 + y2);
    atomicAdd(&tz[i0], z1 + z2);
    atomicAdd(&el[i0], e1 + e2);
    atomicAdd(&dg[i0], 2.0f);
    // vertex i1: neighbors v0, v2; edges e1, e3
    atomicAdd(&tx[i1], x0 + x2);
    atomicAdd(&ty[i1], y0 + y2);
    atomicAdd(&tz[i1], z0 + z2);
    atomicAdd(&el[i1], e1 + e3);
    atomicAdd(&dg[i1], 2.0f);
    // vertex i2: neighbors v0, v1; edges e2, e3
    atomicAdd(&tx[i2], x0 + x1);
    atomicAdd(&ty[i2], y0 + y1);
    atomicAdd(&tz[i2], z0 + z1);
    atomicAdd(&el[i2], e2 + e3);
    atomicAdd(&dg[i2], 2.0f);
}

// ---------------------------------------------------------------------------
// Kernel 2: vertex-parallel loss + WMMA-based mean reduction.
// Each lane computes s_i = |v - lap| * SCALE / avg_edge, scaled by 1/N.
// Wave reduction: D = A(16x4, lane values) x B(4x16, ones) via
// v_wmma_f32_16x16x4_f32 -> per-lane sum of 8 D VGPRs gives the half-wave
// row-sum; + shfl_xor(16) yields the full wave total on every lane.
// ---------------------------------------------------------------------------
__global__ void lap_vertex_loss(const float* __restrict__ verts,
                                const float* __restrict__ acc,
                                float* __restrict__ out,
                                int N, float invN) {
    int i = blockIdx.x * blockDim.x + threadIdx.x;

    float s = 0.0f;
    if (i < N) {
        const float* tx = acc;
        const float* ty = acc + (size_t)N;
        const float* tz = acc + (size_t)2 * N;
        const float* el = acc + (size_t)3 * N;
        const float* dg = acc + (size_t)4 * N;

        float deg = dg[i];
        float rdeg = 1.0f / deg;                 // deg==0 -> inf -> NaN, matches ref 0/0
        float lx = tx[i] * rdeg;
        float ly = ty[i] * rdeg;
        float lz = tz[i] * rdeg;
        float avg_e = el[i] * rdeg;

        float vx = verts[3 * (long long)i + 0];
        float vy = verts[3 * (long long)i + 1];
        float vz = verts[3 * (long long)i + 2];

        float dx = vx - lx, dy = vy - ly, dz = vz - lz;
        float nrm = sqrtf(dx * dx + dy * dy + dz * dz);
        s = nrm * (EDGE_LENGTH_SCALE / avg_e) * invN;
    }

    // ---- wave32 reduction via V_WMMA_F32_16X16X4_F32 ----
    // A layout (32-bit A 16x4): lanes 0-15 hold M=lane with VGPR0=K0,VGPR1=K1;
    // lanes 16-31 hold M=lane-16 with VGPR0=K2,VGPR1=K3. Put value in K0/K2.
    v2f a;
    a.x = s;
    a.y = 0.0f;
    v2f b;
    b.x = 1.0f;
    b.y = 1.0f;
    v8f c = {};
    v8f d = __builtin_amdgcn_wmma_f32_16x16x4_f32(
        /*neg_a=*/false, a, /*neg_b=*/false, b,
        /*c_mod=*/(short)0, c, /*reuse_a=*/false, /*reuse_b=*/false);

    // D layout: VGPR r, lanes 0-15 -> row M=r; lanes 16-31 -> row M=8+r.
    // Rows are replicated across columns, so summing a lane's 8 VGPRs gives
    // the sum of rows 0-7 (lanes <16) or rows 8-15 (lanes >=16).
    float half_sum = d[0] + d[1] + d[2] + d[3] + d[4] + d[5] + d[6] + d[7];
    float wave_sum = half_sum + __shfl_xor(half_sum, 16, 32);

    __shared__ float wsum[8];
    int lane = threadIdx.x & 31;
    int wid = threadIdx.x >> 5;
    if (lane == 0) wsum[wid] = wave_sum;
    __syncthreads();
    if (threadIdx.x == 0) {
        float t = 0.0f;
#pragma unroll
        for (int w = 0; w < 8; ++w) t += wsum[w];
        atomicAdd(out, t);
    }
}

// ---------------------------------------------------------------------------
extern "C" void kernel_launch(void* const* d_in, const int* in_sizes, int n_in,
                              void* d_out, int out_size, void* d_ws, size_t ws_size,
                              hipStream_t stream) {
    const float* verts = (const float*)d_in[0];
    const long long* idx = (const long long*)d_in[1];
    int N = in_sizes[0] / 3;
    int F = in_sizes[1] / 3;

    float* acc = (float*)d_ws;

    // Accumulators and output must be zeroed every launch (harness does not
    // re-poison between graph replays).
    hipMemsetAsync(acc, 0, (size_t)5 * (size_t)N * sizeof(float), stream);
    hipMemsetAsync(d_out, 0, sizeof(float), stream);

    int threads = 256;
    int fblocks = (F + threads - 1) / threads;
    lap_face_scatter<<<fblocks, threads, 0, stream>>>(verts, idx, acc, N, F);

    int vblocks = (N + threads - 1) / threads;
    lap_vertex_loss<<<vblocks, threads, 0, stream>>>(verts, acc, (float*)d_out,
                                                     N, 1.0f / (float)N);
}